// AdaptedLinear_13383118094952
// MI455X (gfx1250) — compile-verified
//
#include <hip/hip_runtime.h>

// Problem constants (from reference)
#define IN_F   2048
#define OUT_F  2048
#define RNK    16
#define HDIM   256
#define BATCH  16

typedef float v2f __attribute__((ext_vector_type(2)));
typedef float v8f __attribute__((ext_vector_type(8)));

// D(16x16,f32) = A(16x4,f32) x B(4x16,f32) + C   — CDNA5 wave32 WMMA
__device__ __forceinline__ v8f wmma4(v2f a, v2f b, v8f c) {
  return __builtin_amdgcn_wmma_f32_16x16x4_f32(
      /*neg_a=*/false, a, /*neg_b=*/false, b,
      /*c_mod=*/(short)0, c, /*reuse_a=*/false, /*reuse_b=*/false);
}

__device__ __forceinline__ v2f ld2(const float* p) {  // 8B-aligned pair load -> global_load_b64
  return *(const v2f*)p;
}

// ---------------------------------------------------------------------------
// Kernel 1: h[b][k] = emb_id[ids[b]][k] + emb_layer[layer_id][k]; zero t.
// ---------------------------------------------------------------------------
__global__ void k_prep(const int* __restrict__ ids, const int* __restrict__ layer_id,
                       const float* __restrict__ emb_id, const float* __restrict__ emb_layer,
                       float* __restrict__ h, float* __restrict__ t) {
  int b = blockIdx.x;          // 16 blocks
  int k = threadIdx.x;         // 256 threads == HDIM
  int id  = ids[b];
  int lid = layer_id[0];
  h[b * HDIM + k] = emb_id[id * HDIM + k] + emb_layer[lid * HDIM + k];
  if (b == 0) t[k] = 0.0f;     // t is 16*16 = 256 floats
}

// ---------------------------------------------------------------------------
// Kernel 2: stream W_A once. Per 16-column tile: Atile = h @ W_A[:,tile]
// via 64x wmma (K=256), then fold with x into t[b,r] (lane-reduced, atomic).
// 1 tile/wave -> 2048 waves for max memory-level parallelism.
// ---------------------------------------------------------------------------
#define HSTRIDE 260   // pad 256 -> 260 floats: 16 lanes hit distinct LDS banks
__global__ void __launch_bounds__(256) k_t(const float* __restrict__ WA,
                                           const float* __restrict__ x,
                                           const float* __restrict__ hbuf,
                                           float* __restrict__ t) {
  __shared__ float sh[BATCH * HSTRIDE];
  for (int idx = threadIdx.x; idx < BATCH * HDIM; idx += blockDim.x) {
    sh[(idx >> 8) * HSTRIDE + (idx & 255)] = hbuf[idx];
  }
  __syncthreads();

  int wave = blockIdx.x * (blockDim.x >> 5) + (threadIdx.x >> 5);
  int lane = threadIdx.x & 31;
  int hh = lane >> 4;          // half-wave: selects K pair {2hh, 2hh+1}
  int lq = lane & 15;          // M (A) / N (B,D) lane index

  int col0 = wave * 16;                    // one 16-col tile per wave
  int r  = col0 >> 11;                     // r-slab (2048 cols per r)
  int i0 = col0 & (IN_F - 1);

  const float* wp = WA + 2 * hh * (RNK * IN_F) + col0 + lq;  // row-major W_A
  const float* hp = sh + lq * HSTRIDE + 2 * hh;
  v8f c = {};
#pragma unroll 8
  for (int k0 = 0; k0 < HDIM; k0 += 4) {
    v2f a, bb;
    a.x = hp[k0];                          // h[M=lq][k0+2hh]   (ds b64 pair)
    a.y = hp[k0 + 1];
    bb.x = wp[k0 * (RNK * IN_F)];          // W_A[k0+2hh][col0+lq] (strided rows)
    bb.y = wp[(k0 + 1) * (RNK * IN_F)];
    c = wmma4(a, bb, c);
  }

  // c[v] = A[b = v+8*hh][col0 + lq]; fold with x, reduce over 16 N-lanes
  float tacc[8];
#pragma unroll
  for (int v = 0; v < 8; ++v) {
    float s = c[v] * x[(v + 8 * hh) * IN_F + i0 + lq];
    s += __shfl_xor(s, 1, 32);
    s += __shfl_xor(s, 2, 32);
    s += __shfl_xor(s, 4, 32);
    s += __shfl_xor(s, 8, 32);
    tacc[v] = s;
  }
  if (lq == 0) {
#pragma unroll
    for (int v = 0; v < 8; ++v)
      atomicAdd(&t[(v + 8 * hh) * RNK + r], tacc[v]);
  }
}

// ---------------------------------------------------------------------------
// Kernel 3: Q[b][h'*16 + r] = h[b][h'] * t[b][r]   ([16,4096], 256 KB, L2-hot)
// ---------------------------------------------------------------------------
__global__ void k_q(const float* __restrict__ hbuf, const float* __restrict__ t,
                    float* __restrict__ q) {
  int idx = blockIdx.x * blockDim.x + threadIdx.x;   // 65536 total
  int b  = idx >> 12;
  int hp = (idx >> 4) & (HDIM - 1);
  int r  = idx & (RNK - 1);
  q[idx] = hbuf[b * HDIM + hp] * t[b * RNK + r];
}

// ---------------------------------------------------------------------------
// Kernel 4: out[b][o] = bias[o]
// ---------------------------------------------------------------------------
__global__ void k_init(const float* __restrict__ bias, float* __restrict__ out) {
  int idx = blockIdx.x * blockDim.x + threadIdx.x;   // 32768 total
  out[idx] = bias[idx & (OUT_F - 1)];
}

// ---------------------------------------------------------------------------
// Kernel 5: out += X @ weight^T. 64 o-tiles x K-split 16 = 1024 waves.
// B element (k,n) = weight[o0+n][kbase+k]: K pair contiguous -> b64/lane.
// ---------------------------------------------------------------------------
#define BASE_KS 16
__global__ void __launch_bounds__(256) k_base(const float* __restrict__ W,
                                              const float* __restrict__ x,
                                              float* __restrict__ out) {
  int wave = blockIdx.x * (blockDim.x >> 5) + (threadIdx.x >> 5);
  int lane = threadIdx.x & 31;
  int hh = lane >> 4, lq = lane & 15;
  int otile = wave >> 4;                       // 128 tiles of 16 outputs
  int kslab = wave & (BASE_KS - 1);
  int o0 = otile * 16;
  int kbase = kslab * (IN_F / BASE_KS);        // 128 per slab

  const float* wp = W + (o0 + lq) * IN_F + kbase + 2 * hh;
  const float* xp = x + lq * IN_F + kbase + 2 * hh;
  v8f c = {};
#pragma unroll 8
  for (int k0 = 0; k0 < IN_F / BASE_KS; k0 += 4) {
    v2f a  = ld2(xp + k0);                     // x[b=lq][...]   (b64, L2-hot)
    v2f bb = ld2(wp + k0);                     // weight stream  (b64)
    c = wmma4(a, bb, c);
  }
#pragma unroll
  for (int v = 0; v < 8; ++v)
    atomicAdd(&out[(v + 8 * hh) * OUT_F + o0 + lq], c[v]);
}

// ---------------------------------------------------------------------------
// Kernel 6: out += Q @ W_B'  where B(k,o) = W_B[k>>4][ o*16 + (k&15) ].
// K = 4096, split 32 -> 4096 waves. K pair (r, r+1) contiguous -> b64/lane.
// ---------------------------------------------------------------------------
#define DELTA_KS 32
__global__ void __launch_bounds__(256) k_delta(const float* __restrict__ WB,
                                               const float* __restrict__ q,
                                               float* __restrict__ out) {
  int wave = blockIdx.x * (blockDim.x >> 5) + (threadIdx.x >> 5);
  int lane = threadIdx.x & 31;
  int hh = lane >> 4, lq = lane & 15;
  int otile = wave >> 5;                       // 128 tiles
  int kslab = wave & (DELTA_KS - 1);
  int o0 = otile * 16;
  int kbase = kslab * ((HDIM * RNK) / DELTA_KS);  // 128 per slab

  const float* qp = q + lq * (HDIM * RNK) + kbase + 2 * hh;
  v8f c = {};
#pragma unroll 8
  for (int k0 = 0; k0 < (HDIM * RNK) / DELTA_KS; k0 += 4) {
    int k  = kbase + k0 + 2 * hh;              // k = h'*16 + r
    int hp = k >> 4;
    int r  = k & 15;                           // r, r+1 stay in-block (k0%4==0)
    v2f a  = ld2(qp + k0);                     // Q[b=lq][k,k+1]      (b64, L2-hot)
    v2f bb = ld2(WB + hp * (OUT_F * RNK) + (o0 + lq) * RNK + r);  // W_B (b64)
    c = wmma4(a, bb, c);
  }
#pragma unroll
  for (int v = 0; v < 8; ++v)
    atomicAdd(&out[(v + 8 * hh) * OUT_F + o0 + lq], c[v]);
}

// ---------------------------------------------------------------------------
extern "C" void kernel_launch(void* const* d_in, const int* in_sizes, int n_in,
                              void* d_out, int out_size, void* d_ws, size_t ws_size,
                              hipStream_t stream) {
  const float* x        = (const float*)d_in[0];
  const int*   ids      = (const int*)  d_in[1];
  const int*   layer_id = (const int*)  d_in[2];
  const float* weight   = (const float*)d_in[3];
  const float* bias     = (const float*)d_in[4];
  const float* emb_id   = (const float*)d_in[5];
  const float* emb_lay  = (const float*)d_in[6];
  const float* W_A      = (const float*)d_in[7];
  const float* W_B      = (const float*)d_in[8];
  float* out = (float*)d_out;

  char* ws = (char*)d_ws;
  float* h = (float*)(ws);             // 16*256 f32 = 16 KB
  float* t = (float*)(ws + 16384);     // 16*16  f32 =  1 KB
  float* q = (float*)(ws + 32768);     // 16*4096 f32 = 256 KB

  k_prep <<<16,  256, 0, stream>>>(ids, layer_id, emb_id, emb_lay, h, t);
  k_t    <<<256, 256, 0, stream>>>(W_A, x, h, t);     // 2048 waves, streams 33.5 MB
  k_q    <<<256, 256, 0, stream>>>(h, t, q);
  k_init <<<128, 256, 0, stream>>>(bias, out);
  k_base <<<128, 256, 0, stream>>>(weight, x, out);   // 1024 waves, streams 8.4 MB
  k_delta<<<512, 256, 0, stream>>>(W_B, q, out);      // 4096 waves, streams 33.5 MB
}